// Bert4Re_67516885893540
// MI455X (gfx1250) — compile-verified
//
#include <hip/hip_runtime.h>

typedef __attribute__((ext_vector_type(2))) float v2f;
typedef __attribute__((ext_vector_type(8))) float v8f;
typedef __attribute__((ext_vector_type(4))) float f4;

#define B_ 32
#define L_ 512
#define D_ 768
#define M_ 24
#define P_ (M_ * (M_ - 1))   // 552 off-diagonal pairs

// ---------------------------------------------------------------------------
// Kernel 1: span-average pooling via V_WMMA_F32_16X16X4_F32.
// One block per (b, m) span. pooled[b,m,d] = (1/len) * sum_{k<len} x[b, head[b,start+k], d].
// Formulated as D = A x B with A row0 = (k<len ? 1/len : 0), B = 4x16 tile of
// gathered rows. 4 chained WMMAs reduce the <=16-row span; 48 column tiles
// split 6-per-wave across 8 waves. Row M=0 of C/D lands in c[0], lanes 0..15.
// ---------------------------------------------------------------------------
__global__ __launch_bounds__(256) void span_pool_wmma(
    const float* __restrict__ seq,       // [B, L, D]
    const int*   __restrict__ heads,     // [B, L]
    const int*   __restrict__ mentions,  // [B, M, 2]
    float*       __restrict__ pooled)    // [B, M, D]  (workspace)
{
  const int bm    = blockIdx.x;          // 0 .. B*M-1
  const int b     = bm / M_;
  const int start = mentions[bm * 2 + 0];
  const int end   = mentions[bm * 2 + 1];
  const int len   = end - start;         // 1 .. 16, start+15 < L guaranteed
  const float inv = 1.0f / (float)len;

  const int lane = threadIdx.x & 31;
  const int wave = threadIdx.x >> 5;
  const int hi2  = (lane >= 16) ? 2 : 0; // K-slot half select (layout-consistent)
  const int n    = lane & 15;            // N / column-within-tile index

  // Per-lane row element offsets and A-matrix values for its 8 K slots
  // (2 VGPR slots per chunk x 4 chunks of K=4).
  const int* hrow = heads + b * L_;
  size_t rowoff[8];
  float  aval[8];
#pragma unroll
  for (int c = 0; c < 4; ++c) {
    const int k0 = c * 4 + hi2;
    const int k1 = k0 + 1;
    rowoff[2 * c]     = ((size_t)b * L_ + (size_t)hrow[start + k0]) * D_;
    rowoff[2 * c + 1] = ((size_t)b * L_ + (size_t)hrow[start + k1]) * D_;
    aval[2 * c]       = (k0 < len) ? inv : 0.0f;   // fold 1/len into A
    aval[2 * c + 1]   = (k1 < len) ? inv : 0.0f;
  }

  // 48 tiles of 16 columns; 8 waves x 6 tiles.
#pragma unroll
  for (int t = 0; t < 6; ++t) {
    const int d0 = (wave * 6 + t) * 16;
    v8f c = {};
#pragma unroll
    for (int ch = 0; ch < 4; ++ch) {
      v2f a, bb;
      a.x  = aval[2 * ch];
      a.y  = aval[2 * ch + 1];
      bb.x = seq[rowoff[2 * ch]     + d0 + n];
      bb.y = seq[rowoff[2 * ch + 1] + d0 + n];
      // 8 args: (neg_a, A, neg_b, B, c_mod, C, reuse_a, reuse_b)
      c = __builtin_amdgcn_wmma_f32_16x16x4_f32(
          false, a, false, bb, (short)0, c, false, false);
    }
    if (lane < 16)
      pooled[(size_t)bm * D_ + d0 + n] = c[0];   // D row M=0: lanes 0-15, N=lane
  }
}

// ---------------------------------------------------------------------------
// Kernel 2: pair expansion writer (the bandwidth-dominant 108.5 MB stream).
// One block per (b, p): concat(pooled[b,i], pooled[b,j]) -> out row of 1536
// floats as 384 float4 NT stores (192 threads x 2), plus 5 pair indices.
// pooled reads are L2-resident (2.36 MB workspace, each row reused 46x).
// ---------------------------------------------------------------------------
__global__ __launch_bounds__(192) void expand_pairs(
    const float* __restrict__ pooled,    // [B, M, D]
    const int*   __restrict__ mentions,  // [B, M, 2]
    float*       __restrict__ out)       // [B*P, 2D] floats, then [B*P, 5] pairs
{
  const int bp = blockIdx.x;             // 0 .. B*P-1
  const int b  = bp / P_;
  const int p  = bp % P_;
  const int i  = p / (M_ - 1);
  const int r  = p % (M_ - 1);
  const int j  = (r < i) ? r : r + 1;    // row-major off-diagonal enumeration

  const f4* obj = (const f4*)(pooled + ((size_t)b * M_ + i) * D_);
  const f4* sub = (const f4*)(pooled + ((size_t)b * M_ + j) * D_);
  f4* dst = (f4*)(out + (size_t)bp * (2 * D_));

  const int t = threadIdx.x;             // 0..191 == D_/4
  __builtin_nontemporal_store(obj[t], dst + t);
  __builtin_nontemporal_store(sub[t], dst + 192 + t);

  if (t < 5) {
    float v;
    switch (t) {
      case 0:  v = (float)b; break;
      case 1:  v = (float)mentions[(b * M_ + i) * 2 + 0]; break;
      case 2:  v = (float)mentions[(b * M_ + i) * 2 + 1]; break;
      case 3:  v = (float)mentions[(b * M_ + j) * 2 + 0]; break;
      default: v = (float)mentions[(b * M_ + j) * 2 + 1]; break;
    }
    out[(size_t)B_ * P_ * 2 * D_ + (size_t)bp * 5 + t] = v;
  }
}

extern "C" void kernel_launch(void* const* d_in, const int* in_sizes, int n_in,
                              void* d_out, int out_size, void* d_ws, size_t ws_size,
                              hipStream_t stream) {
  const float* seq      = (const float*)d_in[0];  // [32,512,768] f32
  const int*   heads    = (const int*)d_in[1];    // [32,512] i32
  const int*   mentions = (const int*)d_in[2];    // [32,24,2] i32
  float* pooled = (float*)d_ws;                   // 32*24*768*4 = 2.36 MB scratch
  float* out    = (float*)d_out;

  span_pool_wmma<<<B_ * M_, 256, 0, stream>>>(seq, heads, mentions, pooled);
  expand_pairs<<<B_ * P_, 192, 0, stream>>>(pooled, mentions, out);
}